// AttnPFN_14585708937405
// MI455X (gfx1250) — compile-verified
//
#include <hip/hip_runtime.h>
#include <math.h>

typedef __attribute__((ext_vector_type(2))) float v2f;
typedef __attribute__((ext_vector_type(8))) float v8f;

#define NBLK 256      // blocks for kernels 1 & 3
#define WPB  8        // waves per block (256 threads, wave32)
#define MPTS 32       // points per pillar

__device__ __forceinline__ float wave_sum32(float v) {
#pragma unroll
  for (int m = 16; m >= 1; m >>= 1) v += __shfl_xor(v, m, 32);
  return v;
}

// ---------------------------------------------------------------------------
// Kernel 1: per-pillar x = feats @ pfn_w via V_WMMA_F32_16X16X4_F32,
// accumulate deterministic per-block partial (sum, sumsq) per channel.
// ---------------------------------------------------------------------------
__global__ __launch_bounds__(256) void pfn_stats_kernel(
    const float* __restrict__ features, const int* __restrict__ num_points,
    const int* __restrict__ coors, const float* __restrict__ pfnw,
    float* __restrict__ partial, int N)
{
  __shared__ float aT[WPB][MPTS * 12];   // per-wave A staging (32 pts x 12 K-padded)
  __shared__ float stage[WPB][128];      // per-wave channel partials (64 sum + 64 sq)

  const int tid  = threadIdx.x;
  const int wave = tid >> 5;
  const int lane = tid & 31;
  const int lh   = lane & 15;
  const bool hi  = lane >= 16;

  // B fragments: pfn_w is 10x64 row-major, K padded to 12 with zeros.
  // B 4x16 f32 layout: VGPR0 = rows {4k, 4k+2}, VGPR1 = rows {4k+1, 4k+3}.
  v2f bf[3][4];
#pragma unroll
  for (int kt = 0; kt < 3; ++kt) {
    const int r0 = 4 * kt + (hi ? 2 : 0);
#pragma unroll
    for (int n = 0; n < 4; ++n) {
      const int col = n * 16 + lh;
      v2f b;
      b.x = (r0     < 10) ? pfnw[r0 * 64 + col]       : 0.0f;
      b.y = (r0 + 1 < 10) ? pfnw[(r0 + 1) * 64 + col] : 0.0f;
      bf[kt][n] = b;
    }
  }

  float s[4]  = {0.0f, 0.0f, 0.0f, 0.0f};
  float sq[4] = {0.0f, 0.0f, 0.0f, 0.0f};

  for (int pil = blockIdx.x * WPB + wave; pil < N; pil += gridDim.x * WPB) {
    // lane == point index within pillar
    const float4 f4 = ((const float4*)features)[pil * MPTS + lane];
    const float x = f4.x, y = f4.y, z = f4.z;
    const int   np  = num_points[pil];
    const float npf = (float)np;
    const float mx = wave_sum32(x) / npf;
    const float my = wave_sum32(y) / npf;
    const float mz = wave_sum32(z) / npf;
    const int c1 = coors[pil * 4 + 1], c2 = coors[pil * 4 + 2], c3 = coors[pil * 4 + 3];
    const float fcx = x - ((float)c3 * 0.2f + 0.1f);
    const float fcy = y - ((float)c2 * 0.2f - 39.9f);
    const float fcz = z - ((float)c1 * 4.0f - 1.0f);
    const float mk  = (lane < np) ? 1.0f : 0.0f;

    float* A = &aT[wave][lane * 12];
    __builtin_amdgcn_wave_barrier();
    A[0] = fcx * mk; A[1] = fcy * mk; A[2] = fcz * mk; A[3] = f4.w * mk;
    A[4] = (x - mx) * mk; A[5] = (y - my) * mk; A[6] = (z - mz) * mk;
    A[7] = fcx * mk; A[8] = fcy * mk; A[9] = fcz * mk;
    A[10] = 0.0f; A[11] = 0.0f;
    __builtin_amdgcn_wave_barrier();   // same-wave DS ops are hardware in-order

    v8f acc[2][4];
#pragma unroll
    for (int m = 0; m < 2; ++m)
#pragma unroll
      for (int n = 0; n < 4; ++n) {
        v8f zz = {0.0f, 0.0f, 0.0f, 0.0f, 0.0f, 0.0f, 0.0f, 0.0f};
        acc[m][n] = zz;
      }

#pragma unroll
    for (int kt = 0; kt < 3; ++kt) {
#pragma unroll
      for (int m = 0; m < 2; ++m) {
        const float* Ap = &aT[wave][(m * 16 + lh) * 12 + 4 * kt + (hi ? 2 : 0)];
        v2f a; a.x = Ap[0]; a.y = Ap[1];
#pragma unroll
        for (int n = 0; n < 4; ++n)
          acc[m][n] = __builtin_amdgcn_wmma_f32_16x16x4_f32(
              false, a, false, bf[kt][n], (short)0, acc[m][n], false, false);
      }
    }
    __builtin_amdgcn_wave_barrier();

    // Per-lane channel partials: lane covers channel n*16 + (lane&15)
#pragma unroll
    for (int n = 0; n < 4; ++n)
#pragma unroll
      for (int m = 0; m < 2; ++m)
#pragma unroll
        for (int r = 0; r < 8; ++r) {
          const float v = acc[m][n][r];
          s[n] += v; sq[n] += v * v;
        }
  }

  // combine the two lane-halves (same channel), stage, fixed-order block reduce
#pragma unroll
  for (int n = 0; n < 4; ++n) {
    s[n]  += __shfl_xor(s[n], 16, 32);
    sq[n] += __shfl_xor(sq[n], 16, 32);
  }
  if (lane < 16) {
#pragma unroll
    for (int n = 0; n < 4; ++n) {
      stage[wave][n * 16 + lane]      = s[n];
      stage[wave][64 + n * 16 + lane] = sq[n];
    }
  }
  __syncthreads();
  if (tid < 128) {
    float t = 0.0f;
#pragma unroll
    for (int w = 0; w < WPB; ++w) t += stage[w][tid];
    partial[blockIdx.x * 128 + tid] = t;   // written unconditionally every call
  }
}

// ---------------------------------------------------------------------------
// Kernel 2: fold BN stats into per-channel scale/shift (deterministic order).
// ---------------------------------------------------------------------------
__global__ __launch_bounds__(64) void bn_finalize_kernel(
    const float* __restrict__ partial, const float* __restrict__ gamma,
    const float* __restrict__ beta, float* __restrict__ ss, int nblocks, float cnt)
{
  const int c = threadIdx.x;   // 64 channels
  float s = 0.0f, q = 0.0f;
  for (int b = 0; b < nblocks; ++b) {
    s += partial[b * 128 + c];
    q += partial[b * 128 + 64 + c];
  }
  const float mean = s / cnt;
  const float var  = q / cnt - mean * mean;
  const float sc   = rsqrtf(var + 1e-3f) * gamma[c];
  ss[c]      = sc;
  ss[64 + c] = beta[c] - mean * sc;
}

// ---------------------------------------------------------------------------
// Kernel 3: recompute x via WMMA, BN + leaky-ReLU, per-pillar attention,
// fa/fm reduction, write output (N,64).
// ---------------------------------------------------------------------------
__global__ __launch_bounds__(256) void pfn_out_kernel(
    const float* __restrict__ features, const int* __restrict__ num_points,
    const int* __restrict__ coors, const float* __restrict__ pfnw,
    const float* __restrict__ ss,
    const float* __restrict__ wq, const float* __restrict__ wk,
    const float* __restrict__ wv, const float* __restrict__ wo,
    const float* __restrict__ bq, const float* __restrict__ bk,
    const float* __restrict__ bv, const float* __restrict__ bo,
    float* __restrict__ out, int N)
{
  __shared__ float aT[WPB][MPTS * 12];
  __shared__ float kvT[WPB][MPTS * 6];   // per point: k[3], v[3]
  __shared__ float maT[WPB][MPTS];       // per point: max_attn

  const int tid  = threadIdx.x;
  const int wave = tid >> 5;
  const int lane = tid & 31;
  const int lh   = lane & 15;
  const bool hi  = lane >= 16;

  // B fragments (as in kernel 1)
  v2f bf[3][4];
#pragma unroll
  for (int kt = 0; kt < 3; ++kt) {
    const int r0 = 4 * kt + (hi ? 2 : 0);
#pragma unroll
    for (int n = 0; n < 4; ++n) {
      const int col = n * 16 + lh;
      v2f b;
      b.x = (r0     < 10) ? pfnw[r0 * 64 + col]       : 0.0f;
      b.y = (r0 + 1 < 10) ? pfnw[(r0 + 1) * 64 + col] : 0.0f;
      bf[kt][n] = b;
    }
  }

  // BN scale/shift for this lane's channels (channel = n*16 + lane%16)
  float scl[4], shf[4];
#pragma unroll
  for (int n = 0; n < 4; ++n) {
    scl[n] = ss[n * 16 + lh];
    shf[n] = ss[64 + n * 16 + lh];
  }

  // attention weights (uniform, kept in registers)
  float Wq[9], Wk[9], Wv[9], Wo[9], Bq[3], Bk[3], Bv[3], Bo[3];
#pragma unroll
  for (int i = 0; i < 9; ++i) { Wq[i] = wq[i]; Wk[i] = wk[i]; Wv[i] = wv[i]; Wo[i] = wo[i]; }
#pragma unroll
  for (int i = 0; i < 3; ++i) { Bq[i] = bq[i]; Bk[i] = bk[i]; Bv[i] = bv[i]; Bo[i] = bo[i]; }

  for (int pil = blockIdx.x * WPB + wave; pil < N; pil += gridDim.x * WPB) {
    const float4 f4 = ((const float4*)features)[pil * MPTS + lane];
    const float x = f4.x, y = f4.y, z = f4.z;
    const int   np  = num_points[pil];
    const float npf = (float)np;
    const float mx = wave_sum32(x) / npf;
    const float my = wave_sum32(y) / npf;
    const float mz = wave_sum32(z) / npf;
    const int c1 = coors[pil * 4 + 1], c2 = coors[pil * 4 + 2], c3 = coors[pil * 4 + 3];
    const float fcx = x - ((float)c3 * 0.2f + 0.1f);
    const float fcy = y - ((float)c2 * 0.2f - 39.9f);
    const float fcz = z - ((float)c1 * 4.0f - 1.0f);
    const float mk  = (lane < np) ? 1.0f : 0.0f;

    float* A = &aT[wave][lane * 12];
    __builtin_amdgcn_wave_barrier();
    A[0] = fcx * mk; A[1] = fcy * mk; A[2] = fcz * mk; A[3] = f4.w * mk;
    A[4] = (x - mx) * mk; A[5] = (y - my) * mk; A[6] = (z - mz) * mk;
    A[7] = fcx * mk; A[8] = fcy * mk; A[9] = fcz * mk;
    A[10] = 0.0f; A[11] = 0.0f;
    __builtin_amdgcn_wave_barrier();

    v8f acc[2][4];
#pragma unroll
    for (int m = 0; m < 2; ++m)
#pragma unroll
      for (int n = 0; n < 4; ++n) {
        v8f zz = {0.0f, 0.0f, 0.0f, 0.0f, 0.0f, 0.0f, 0.0f, 0.0f};
        acc[m][n] = zz;
      }

#pragma unroll
    for (int kt = 0; kt < 3; ++kt) {
#pragma unroll
      for (int m = 0; m < 2; ++m) {
        const float* Ap = &aT[wave][(m * 16 + lh) * 12 + 4 * kt + (hi ? 2 : 0)];
        v2f a; a.x = Ap[0]; a.y = Ap[1];
#pragma unroll
        for (int n = 0; n < 4; ++n)
          acc[m][n] = __builtin_amdgcn_wmma_f32_16x16x4_f32(
              false, a, false, bf[kt][n], (short)0, acc[m][n], false, false);
      }
    }

    // ---- per-pillar attention on raw xyz (lane == point i) ----
    float qh[3], kh[3], vh[3];
#pragma unroll
    for (int h = 0; h < 3; ++h) {
      qh[h] = x * Wq[h] + y * Wq[3 + h] + z * Wq[6 + h] + Bq[h];
      kh[h] = x * Wk[h] + y * Wk[3 + h] + z * Wk[6 + h] + Bk[h];
      vh[h] = x * Wv[h] + y * Wv[3 + h] + z * Wv[6 + h] + Bv[h];
    }
    __builtin_amdgcn_wave_barrier();
#pragma unroll
    for (int h = 0; h < 3; ++h) {
      kvT[wave][lane * 6 + h]     = kh[h];
      kvT[wave][lane * 6 + 3 + h] = vh[h];
    }
    __builtin_amdgcn_wave_barrier();

    const float* kvp = &kvT[wave][0];
    float mxh[3] = {-3.0e38f, -3.0e38f, -3.0e38f};
#pragma unroll 4
    for (int j = 0; j < MPTS; ++j)
#pragma unroll
      for (int h = 0; h < 3; ++h)
        mxh[h] = fmaxf(mxh[h], qh[h] * kvp[j * 6 + h]);

    float se[3] = {0.0f, 0.0f, 0.0f}, oo[3] = {0.0f, 0.0f, 0.0f};
#pragma unroll 4
    for (int j = 0; j < MPTS; ++j)
#pragma unroll
      for (int h = 0; h < 3; ++h) {
        const float e = expf(qh[h] * kvp[j * 6 + h] - mxh[h]);
        se[h] += e;
        oo[h] += e * kvp[j * 6 + 3 + h];
      }

    float oh[3];
#pragma unroll
    for (int h = 0; h < 3; ++h) oh[h] = oo[h] / se[h];
    float ma = -3.0e38f;
#pragma unroll
    for (int h2 = 0; h2 < 3; ++h2) {
      const float ao = oh[0] * Wo[h2] + oh[1] * Wo[3 + h2] + oh[2] * Wo[6 + h2] + Bo[h2];
      ma = fmaxf(ma, ao);
    }
    maT[wave][lane] = ma;
    const float sum_attn = wave_sum32(ma);
    const float inv_sa = 1.0f / sum_attn;
    __builtin_amdgcn_wave_barrier();

    // max_attn for the 16 points this lane's accumulator rows cover
    float maP[2][8];
#pragma unroll
    for (int m = 0; m < 2; ++m)
#pragma unroll
      for (int r = 0; r < 8; ++r)
        maP[m][r] = maT[wave][m * 16 + (hi ? 8 : 0) + r];
    __builtin_amdgcn_wave_barrier();

    // BN + leaky-ReLU + fa/fm point-max reduction, then output
#pragma unroll
    for (int n = 0; n < 4; ++n) {
      float fmv = -3.0e38f, fav = -3.0e38f;
#pragma unroll
      for (int m = 0; m < 2; ++m)
#pragma unroll
        for (int r = 0; r < 8; ++r) {
          float xv = acc[m][n][r] * scl[n] + shf[n];
          xv = (xv > 0.0f) ? xv : 0.01f * xv;
          fmv = fmaxf(fmv, xv);
          fav = fmaxf(fav, maP[m][r] * xv * inv_sa);
        }
      fmv = fmaxf(fmv, __shfl_xor(fmv, 16, 32));
      fav = fmaxf(fav, __shfl_xor(fav, 16, 32));
      if (lane < 16) out[pil * 64 + n * 16 + lane] = 0.5f * (fav + fmv);
    }
  }
}

// ---------------------------------------------------------------------------
extern "C" void kernel_launch(void* const* d_in, const int* in_sizes, int n_in,
                              void* d_out, int out_size, void* d_ws, size_t ws_size,
                              hipStream_t stream) {
  (void)n_in; (void)out_size; (void)ws_size;
  const float* features   = (const float*)d_in[0];
  const int*   num_points = (const int*)d_in[1];
  const int*   coors      = (const int*)d_in[2];
  const float* pfn_w      = (const float*)d_in[3];
  const float* bn_gamma   = (const float*)d_in[4];
  const float* bn_beta    = (const float*)d_in[5];
  const float* wq = (const float*)d_in[6];
  const float* wk = (const float*)d_in[7];
  const float* wv = (const float*)d_in[8];
  const float* wo = (const float*)d_in[9];
  const float* bq = (const float*)d_in[10];
  const float* bk = (const float*)d_in[11];
  const float* bv = (const float*)d_in[12];
  const float* bo = (const float*)d_in[13];
  const int N = in_sizes[1];             // num_points has N elements

  float* ws      = (float*)d_ws;
  float* partial = ws;                   // NBLK * 128 floats
  float* ss      = ws + NBLK * 128;      // 64 scale + 64 shift

  pfn_stats_kernel<<<NBLK, 256, 0, stream>>>(features, num_points, coors, pfn_w,
                                             partial, N);
  bn_finalize_kernel<<<1, 64, 0, stream>>>(partial, bn_gamma, bn_beta, ss, NBLK,
                                           (float)N * 32.0f);
  pfn_out_kernel<<<NBLK, 256, 0, stream>>>(features, num_points, coors, pfn_w, ss,
                                           wq, wk, wv, wo, bq, bk, bv, bo,
                                           (float*)d_out, N);
}